// DistMul_88519275970866
// MI455X (gfx1250) — compile-verified
//
#include <hip/hip_runtime.h>
#include <hip/hip_bf16.h>
#include <stdint.h>

#define EMB_DIM    128
#define TOTAL_REL  500
#define REL_LDS_BYTES (TOTAL_REL * EMB_DIM * 4)   // 256000 B <= 320 KB WGP LDS

typedef __attribute__((ext_vector_type(4))) unsigned int u32x4;
typedef __attribute__((ext_vector_type(8))) int          i32x8;
typedef __attribute__((ext_vector_type(4))) int          i32x4;

__global__ __launch_bounds__(1024)
void distmult_kernel(const int* __restrict__ bh,
                     const int* __restrict__ bt,
                     const int* __restrict__ br,
                     const float* __restrict__ ent,
                     const float* __restrict__ rel,
                     float* __restrict__ out, int n)
{
    extern __shared__ float lds_rel[];   // 500 x 128 f32, LDS offset 0

#if __has_builtin(__builtin_amdgcn_tensor_load_to_lds) && __has_builtin(__builtin_amdgcn_s_wait_tensorcnt)
    // ---- Stage the whole relation table into LDS with one TDM descriptor ----
    if (threadIdx.x < 32) {              // one wave issues; TDM ignores EXEC
        unsigned long long ga = (unsigned long long)(uintptr_t)rel;
        u32x4 g0;
        g0[0] = 1u;                                   // count=1, user D#, no gather
        g0[1] = 0u;                                   // lds_addr = 0 (dynamic LDS base)
        g0[2] = (unsigned)(ga & 0xFFFFFFFFu);         // global_addr[31:0]
        g0[3] = (unsigned)((ga >> 32) & 0x1FFFFFFu)   // global_addr[56:32]
              | (2u << 30);                           // type = 2 ("image")
        i32x8 g1;
        g1[0] = (int)(2u << 16);                      // wg_mask=0, data_size=4B
        g1[1] = (int)((unsigned)EMB_DIM << 16);       // tensor_dim0 = 128
        g1[2] = (int)((unsigned)TOTAL_REL << 16);     // tensor_dim1 = 500 (lo16)
        g1[3] = (int)((unsigned)EMB_DIM << 16);       // tile_dim0 = 128
        g1[4] = (int)TOTAL_REL;                       // tile_dim1 = 500, tile_dim2 = 0
        g1[5] = (int)EMB_DIM;                         // tensor_dim0_stride = 128
        g1[6] = (int)((unsigned)((EMB_DIM * TOTAL_REL) & 0xFFFF) << 16); // dim1_stride lo16
        g1[7] = (int)((unsigned)(EMB_DIM * TOTAL_REL) >> 16);            // dim1_stride hi
        i32x4 gz4 = {0, 0, 0, 0};
        i32x8 gz8 = {0, 0, 0, 0, 0, 0, 0, 0};
        __builtin_amdgcn_tensor_load_to_lds(g0, g1, gz4, gz4, gz8, 0);
        __builtin_amdgcn_s_wait_tensorcnt(0);
    }
#else
    // Fallback: cooperative vector copy
    for (int idx = threadIdx.x; idx < TOTAL_REL * EMB_DIM / 4; idx += blockDim.x)
        ((float4*)lds_rel)[idx] = ((const float4*)rel)[idx];
#endif
    __syncthreads();

    const int lane = threadIdx.x & 31;
    const int sub  = lane & 15;          // position within half-wave
    const int half = lane >> 4;          // which triple of the pair
    const int wid  = blockIdx.x * (blockDim.x >> 5) + (threadIdx.x >> 5);
    const int nw   = gridDim.x * (blockDim.x >> 5);

    const float4* __restrict__ entv = (const float4*)ent;
    const float4* __restrict__ relv = (const float4*)lds_rel;

    // Two triples per wave: 16 lanes x 2 float4 = one 512 B row per half-wave.
    for (int p = wid; 2 * p < n; p += nw) {
        const int iraw = 2 * p + half;
        const int i    = (iraw < n) ? iraw : (n - 1);   // clamp (n is even anyway)

        const long hbase = (long)bh[i] * (EMB_DIM / 4);
        const long tbase = (long)bt[i] * (EMB_DIM / 4);
        const int  ridx  = br[i];

        // Prefetch the next pair's rows: 32 lanes at stride-2 float4 covers
        // all four 128 B lines of both halves' 512 B rows per instruction.
        const int pn = p + nw;
        if (2 * pn < n) {
            const int inx = 2 * pn + half;
            const int inc = (inx < n) ? inx : (n - 1);
            const long hn = (long)bh[inc] * (EMB_DIM / 4);
            const long tn = (long)bt[inc] * (EMB_DIM / 4);
            __builtin_prefetch(&entv[hn + 2 * sub], 0, 3);
            __builtin_prefetch(&entv[tn + 2 * sub], 0, 3);
        }

        // Issue all four 512 B gather loads before any reduction (max MLP).
        const float4 hv0 = entv[hbase + sub];
        const float4 hv1 = entv[hbase + sub + 16];
        const float4 tv0 = entv[tbase + sub];
        const float4 tv1 = entv[tbase + sub + 16];
        const float4 rv0 = relv[ridx * (EMB_DIM / 4) + sub];
        const float4 rv1 = relv[ridx * (EMB_DIM / 4) + sub + 16];

        float s = fmaf(hv0.x * rv0.x, tv0.x,
                  fmaf(hv0.y * rv0.y, tv0.y,
                  fmaf(hv0.z * rv0.z, tv0.z,
                  fmaf(hv0.w * rv0.w, tv0.w,
                  fmaf(hv1.x * rv1.x, tv1.x,
                  fmaf(hv1.y * rv1.y, tv1.y,
                  fmaf(hv1.z * rv1.z, tv1.z, hv1.w * rv1.w * tv1.w)))))));

        // 4-step butterfly within each 16-lane half (2 triples reduced at once)
        #pragma unroll
        for (int off = 8; off > 0; off >>= 1)
            s += __shfl_xor(s, off, 32);   // xor<=8 never crosses the halves

        if (sub == 0 && iraw < n)
            out[i] = 1.0f / (1.0f + __expf(-s));   // lanes 0,16 -> adjacent stores
    }
}

extern "C" void kernel_launch(void* const* d_in, const int* in_sizes, int n_in,
                              void* d_out, int out_size, void* d_ws, size_t ws_size,
                              hipStream_t stream) {
    (void)n_in; (void)out_size; (void)d_ws; (void)ws_size;
    const int*   bh  = (const int*)d_in[0];
    const int*   bt  = (const int*)d_in[1];
    const int*   br  = (const int*)d_in[2];
    const float* ent = (const float*)d_in[3];
    const float* rel = (const float*)d_in[4];
    float* out = (float*)d_out;
    const int n = in_sizes[0];

    static const int lds_attr_set = []() {
        return (int)hipFuncSetAttribute((const void*)distmult_kernel,
                                        hipFuncAttributeMaxDynamicSharedMemorySize,
                                        REL_LDS_BYTES);
    }();
    (void)lds_attr_set;

    const int blocks = 1024;   // 32768 waves, 2 triples/wave/iter -> 16 iters/wave
    distmult_kernel<<<blocks, 1024, REL_LDS_BYTES, stream>>>(bh, bt, br, ent, rel, out, n);
}